// OtherRNN_81174881894477
// MI455X (gfx1250) — compile-verified
//
#include <hip/hip_runtime.h>

typedef __attribute__((ext_vector_type(2))) float v2f;
typedef __attribute__((ext_vector_type(8))) float v8f;

#define HID  10
#define OBSN 16
#define INPN 19      // OBS + ACT
#define LSEQ 512
#define LDSP 18      // padded LDS row pitch (floats): keeps b64 align, spreads banks

static __device__ __forceinline__ v8f wmma4(v2f a, v2f b, v8f c) {
    // V_WMMA_F32_16X16X4_F32 : D = A(16x4,f32) * B(4x16,f32) + C(16x16,f32)
    return __builtin_amdgcn_wmma_f32_16x16x4_f32(
        /*neg_a=*/false, a, /*neg_b=*/false, b,
        /*c_mod=*/(short)0, c, /*reuse_a=*/false, /*reuse_b=*/false);
}

__global__ __launch_bounds__(32)
void rnn_fused_wmma(const float* __restrict__ x,
                    const float* __restrict__ Wg, const float* __restrict__ bg,
                    const float* __restrict__ Wr, const float* __restrict__ br,
                    const float* __restrict__ W1, const float* __restrict__ b1,
                    const float* __restrict__ W2, const float* __restrict__ b2,
                    float* __restrict__ out)
{
    __shared__ float hT[16 * LDSP];   // h tile: row = batch-in-tile, col = hidden
    __shared__ float zT[16 * LDSP];   // relu scratch tile

    const int  lane = threadIdx.x;          // 0..31 (wave32)
    const int  col  = lane & 15;            // N-column for B/C/D operands; also A row (M)
    const bool hi   = lane >= 16;
    const int  krow = hi ? 2 : 0;           // f32 A/B pair layout: v0=K|K+2, v1=K+1|K+3
    const int  rofs = hi ? 8 : 0;           // D layout: vgpr r -> row r (lo) / r+8 (hi)
    const int  b0   = blockIdx.x * 16;

    // ---------------- preload weights in B-operand layout ----------------
    v2f Bg[4], Bx[5], Bh[3], B1m[3], B2m[3];
    #pragma unroll
    for (int p = 0; p < 4; ++p) {           // Wg: 16x10, K=16 exact
        int k = 4 * p + krow;
        Bg[p].x = (col < HID) ? Wg[k * HID + col]       : 0.f;
        Bg[p].y = (col < HID) ? Wg[(k + 1) * HID + col] : 0.f;
    }
    #pragma unroll
    for (int p = 0; p < 5; ++p) {           // Wr rows 0..18 (x part), K pad 19->20
        int k = 4 * p + krow;
        Bx[p].x = (k     < INPN && col < HID) ? Wr[k * HID + col]       : 0.f;
        Bx[p].y = (k + 1 < INPN && col < HID) ? Wr[(k + 1) * HID + col] : 0.f;
    }
    #pragma unroll
    for (int p = 0; p < 3; ++p) {           // Wr rows 19..28 (h part), W1, W2 : K pad 10->12
        int k = 4 * p + krow;
        Bh[p].x  = (k     < HID && col < HID) ? Wr[(INPN + k)     * HID + col] : 0.f;
        Bh[p].y  = (k + 1 < HID && col < HID) ? Wr[(INPN + k + 1) * HID + col] : 0.f;
        B1m[p].x = (k     < HID && col < HID) ? W1[k * HID + col]       : 0.f;
        B1m[p].y = (k + 1 < HID && col < HID) ? W1[(k + 1) * HID + col] : 0.f;
        B2m[p].x = (k     < HID) ? W2[k * OBSN + col]       : 0.f;
        B2m[p].y = (k + 1 < HID) ? W2[(k + 1) * OBSN + col] : 0.f;
    }

    // ---------------- bias-broadcast C initializers ----------------
    v8f cR0 = {}, c10 = {}, c20 = {}, cG0 = {};
    {
        float vr = (col < HID) ? br[col] : 0.f;
        float v1 = (col < HID) ? b1[col] : 0.f;
        float v2 = b2[col];
        float vg = (col < HID) ? bg[col] : 0.f;
        #pragma unroll
        for (int i = 0; i < 8; ++i) { cR0[i] = vr; c10[i] = v1; c20[i] = v2; cG0[i] = vg; }
    }

    // A-operand row of x for this lane (M = col = batch row within tile)
    const float* xrow = x + (size_t)(b0 + col) * LSEQ * INPN;

    // ---------------- h0 = x[:,0,0:16] @ Wg + bg ----------------
    {
        v8f d = cG0;
        #pragma unroll
        for (int p = 0; p < 4; ++p) {
            int k = 4 * p + krow;
            v2f a; a.x = xrow[k]; a.y = xrow[k + 1];
            d = wmma4(a, Bg[p], d);
        }
        #pragma unroll
        for (int r = 0; r < 8; ++r) hT[(r + rofs) * LDSP + col] = d[r];
    }

    // ---------------- sequential scan over L ----------------
    for (int t = 0; t < LSEQ; ++t) {
        const float* xr = xrow + t * INPN;
        if (t + 1 < LSEQ) __builtin_prefetch(xr + INPN, 0, 1);   // global_prefetch_b8

        // gather x_t A-tile (off critical chain)
        v2f ax[5];
        #pragma unroll
        for (int p = 0; p < 5; ++p) {
            int k = 4 * p + krow;
            ax[p].x = (k     < INPN) ? xr[k]     : 0.f;
            ax[p].y = (k + 1 < INPN) ? xr[k + 1] : 0.f;
        }

        // h_new = [x_t, h] @ Wr + br   (x part: 5 WMMA; recurrent part: 3 WMMA)
        v8f d = cR0;
        #pragma unroll
        for (int p = 0; p < 5; ++p) d = wmma4(ax[p], Bx[p], d);
        #pragma unroll
        for (int p = 0; p < 3; ++p) {
            int k = 4 * p + krow;
            v2f a; a.x = hT[col * LDSP + k]; a.y = hT[col * LDSP + k + 1];
            d = wmma4(a, Bh[p], d);
        }

        // publish h_new (D layout -> LDS row-major) for A-format reloads
        #pragma unroll
        for (int r = 0; r < 8; ++r) hT[(r + rofs) * LDSP + col] = d[r];

        // z = relu(h_new @ W1 + b1)
        v8f z = c10;
        #pragma unroll
        for (int p = 0; p < 3; ++p) {
            int k = 4 * p + krow;
            v2f a; a.x = hT[col * LDSP + k]; a.y = hT[col * LDSP + k + 1];
            z = wmma4(a, B1m[p], z);
        }
        #pragma unroll
        for (int i = 0; i < 8; ++i) z[i] = fmaxf(z[i], 0.f);
        #pragma unroll
        for (int r = 0; r < 8; ++r) zT[(r + rofs) * LDSP + col] = z[r];

        // o = z @ W2 + b2
        v8f o = c20;
        #pragma unroll
        for (int p = 0; p < 3; ++p) {
            int k = 4 * p + krow;
            v2f a; a.x = zT[col * LDSP + k]; a.y = zT[col * LDSP + k + 1];
            o = wmma4(a, B2m[p], o);
        }

        // store o: D row r -> batch b0+r(+8 for hi half), col = obs index
        #pragma unroll
        for (int r = 0; r < 8; ++r)
            out[((size_t)(b0 + r + rofs) * LSEQ + t) * OBSN + col] = o[r];
    }
}

extern "C" void kernel_launch(void* const* d_in, const int* in_sizes, int n_in,
                              void* d_out, int out_size, void* d_ws, size_t ws_size,
                              hipStream_t stream) {
    const float* x  = (const float*)d_in[0];
    // d_in[1] = dt (unused by the reference forward)
    const float* Wg = (const float*)d_in[2];
    const float* bg = (const float*)d_in[3];
    const float* Wr = (const float*)d_in[4];
    const float* br = (const float*)d_in[5];
    const float* W1 = (const float*)d_in[6];
    const float* b1 = (const float*)d_in[7];
    const float* W2 = (const float*)d_in[8];
    const float* b2 = (const float*)d_in[9];
    float* out = (float*)d_out;

    const int B = in_sizes[0] / (LSEQ * INPN);   // 4096
    dim3 grid(B / 16), block(32);
    rnn_fused_wmma<<<grid, block, 0, stream>>>(x, Wg, bg, Wr, br, W1, b1, W2, b2, out);
}